// EquilibriumModel_83416854823069
// MI455X (gfx1250) — compile-verified
//
#include <hip/hip_runtime.h>
#include <math.h>

#define N_TRAILS 256
#define N_STEPS  16
#define N_NODES  4096   // N_TRAILS*N_STEPS
#define N_EDGES  7920
#define N_TRAIL_E 3840

typedef float v2f __attribute__((ext_vector_type(2)));
typedef float v8f __attribute__((ext_vector_type(8)));

// ---------------------------------------------------------------------------
// K1: recover edge endpoints (u,v) from connectivity C (one -1 and one +1 per
// row) with the matrix engine:  D = C_tile @ B,  B cols = [n, n^2] (exact in
// f32: 4095^2 < 2^24, C in {0,+-1}).  a = v-u, b = v^2-u^2  =>  v+u = b/a.
// One wave per 16-edge tile; K loop over 4096 in steps of 8 (two WMMA chains).
// ---------------------------------------------------------------------------
__global__ __launch_bounds__(32) void cem_edges_wmma(const float* __restrict__ C,
                                                     int* __restrict__ eu,
                                                     int* __restrict__ ev) {
  const int lane  = threadIdx.x;         // 0..31
  const int e0    = blockIdx.x * 16;
  const int m     = lane & 15;           // A row / D row-half
  const int khalf = (lane >> 4) << 1;    // 0 or 2: A K-offset within 4-chunk
  const int ncol  = m;                   // B column held by this lane

  const float* rowp = C + (size_t)(e0 + m) * (size_t)N_NODES + khalf;

  v8f acc0 = {}; v8f acc1 = {};
  for (int k = 0; k < N_NODES; k += 8) {
    __builtin_prefetch(rowp + k + 64, 0, 1);          // global_prefetch_b8
    v2f a0 = *(const v2f*)(rowp + k);                 // cols k+khalf, +1
    v2f a1 = *(const v2f*)(rowp + k + 4);             // cols k+4+khalf, +1
    float kg0 = (float)(k + khalf),     kg0b = kg0 + 1.0f;
    float kg1 = (float)(k + 4 + khalf), kg1b = kg1 + 1.0f;
    v2f b0, b1;
    b0.x = (ncol == 0) ? kg0  : ((ncol == 1) ? kg0  * kg0  : 0.0f);
    b0.y = (ncol == 0) ? kg0b : ((ncol == 1) ? kg0b * kg0b : 0.0f);
    b1.x = (ncol == 0) ? kg1  : ((ncol == 1) ? kg1  * kg1  : 0.0f);
    b1.y = (ncol == 0) ? kg1b : ((ncol == 1) ? kg1b * kg1b : 0.0f);
    acc0 = __builtin_amdgcn_wmma_f32_16x16x4_f32(false, a0, false, b0,
                                                 (short)0, acc0, false, false);
    acc1 = __builtin_amdgcn_wmma_f32_16x16x4_f32(false, a1, false, b1,
                                                 (short)0, acc1, false, false);
  }

  __shared__ float dtile[32][8];
  #pragma unroll
  for (int r = 0; r < 8; ++r) dtile[lane][r] = acc0[r] + acc1[r];
  __syncthreads();

  if (lane < 16) {
    // D layout: D[m][n] -> VGPR (m&7), lane n (m<8) or 16+n (m>=8)
    int base = (lane < 8) ? 0 : 16;
    float a  = dtile[base + 0][lane & 7];   // col 0: v - u
    float bq = dtile[base + 1][lane & 7];   // col 1: v^2 - u^2
    int ai = (int)lrintf(a);
    int si = (int)lrintf(bq / a);           // v + u (exact integer ratio)
    eu[e0 + lane] = (si - ai) / 2;
    ev[e0 + lane] = (si + ai) / 2;
  }
}

// ---------------------------------------------------------------------------
// K2: the whole 16-step scan in ONE workgroup (256 threads = 8 waves).
// xyz state (48KB), inverse node map (8KB) and deviation accumulators (3KB)
// live in LDS; deviation scatter uses ds_add_f32 atomics.
// ---------------------------------------------------------------------------
__global__ __launch_bounds__(256) void cem_scan(
    const float* __restrict__ xyz0,     const float* __restrict__ loads,
    const float* __restrict__ lengths0, const float* __restrict__ planes,
    const float* __restrict__ forces0,  const float* __restrict__ devmask,
    const int*   __restrict__ seqs,     const int*   __restrict__ seq_edges,
    const int*   __restrict__ eu,       const int*   __restrict__ ev,
    float* __restrict__ out_xyz,   float* __restrict__ out_react,
    float* __restrict__ out_len,   float* __restrict__ out_loads,
    float* __restrict__ out_forces) {
  __shared__ float          xyz_s[N_NODES * 3];   // 48 KB
  __shared__ unsigned short inv_s[N_NODES];       // 8 KB  packed (step<<8)|pos
  __shared__ float          dev_s[N_TRAILS * 3];  // 3 KB
  const int s = threadIdx.x;

  // ---- init LDS + prefill global outputs ----
  for (int n = s; n < N_NODES; n += N_TRAILS) {
    xyz_s[n*3+0] = 0.0f; xyz_s[n*3+1] = 0.0f; xyz_s[n*3+2] = 0.0f;
    inv_s[n] = 0xFFFFu;
    out_react[n*3+0] = 0.0f; out_react[n*3+1] = 0.0f; out_react[n*3+2] = 0.0f;
  }
  __syncthreads();
  for (int k = s; k < N_NODES; k += N_TRAILS) {
    int node = seqs[k];
    if (node >= 0) inv_s[node] = (unsigned short)k;  // k = step*256 + pos
  }
  for (int idx = s; idx < N_NODES * 3; idx += N_TRAILS) out_loads[idx] = loads[idx];
  for (int e = s; e < N_EDGES; e += N_TRAILS) {
    float f = forces0[e];
    out_forces[e] = (f == 0.0f) ? 0.0f : f;   // where(forces0!=0, forces0, 0)
  }
  __threadfence();     // make prefills device-visible before in-scan scatters
  __syncthreads();

  // ---- scan carry: position p (xyz_next chain) and residual r ----
  float px = 0.f, py = 0.f, pz = 0.f, rx = 0.f, ry = 0.f, rz = 0.f;
  { int n0 = seqs[s];
    if (n0 >= 0) { px = xyz0[n0*3+0]; py = xyz0[n0*3+1]; pz = xyz0[n0*3+2]; } }

  for (int i = 0; i < N_STEPS; ++i) {
    const int node = seqs[i * N_TRAILS + s];
    if (node >= 0) {
      xyz_s[node*3+0] = px; xyz_s[node*3+1] = py; xyz_s[node*3+2] = pz;
    }
    dev_s[s*3+0] = 0.0f; dev_s[s*3+1] = 0.0f; dev_s[s*3+2] = 0.0f;
    __syncthreads();

    // deviation scatter: only deviation edges incident to step-i nodes matter
    for (int e = s; e < N_EDGES; e += N_TRAILS) {
      if (devmask[e] != 0.0f) {
        int ue = eu[e], ve = ev[e];
        float dx = xyz_s[ve*3+0] - xyz_s[ue*3+0];
        float dy = xyz_s[ve*3+1] - xyz_s[ue*3+1];
        float dz = xyz_s[ve*3+2] - xyz_s[ue*3+2];
        float s2 = dx*dx + dy*dy + dz*dz;
        float invn = rsqrtf((s2 == 0.0f) ? 1.0f : s2);   // safe_normalize
        float F  = forces0[e] * invn;
        float fx = F * dx, fy = F * dy, fz = F * dz;
        unsigned int pu = inv_s[ue];                      // inc[e,u] = +1
        if ((pu >> 8) == (unsigned)i) {
          int p = pu & 255;
          atomicAdd(&dev_s[p*3+0], fx); atomicAdd(&dev_s[p*3+1], fy);
          atomicAdd(&dev_s[p*3+2], fz);
        }
        unsigned int pv = inv_s[ve];                      // inc[e,v] = -1
        if ((pv >> 8) == (unsigned)i) {
          int p = pv & 255;
          atomicAdd(&dev_s[p*3+0], -fx); atomicAdd(&dev_s[p*3+1], -fy);
          atomicAdd(&dev_s[p*3+2], -fz);
        }
      }
    }
    __syncthreads();

    float L = 0.0f;
    if (node >= 0) {
      rx = rx - dev_s[s*3+0] - loads[node*3+0];
      ry = ry - dev_s[s*3+1] - loads[node*3+1];
      rz = rz - dev_s[s*3+2] - loads[node*3+2];

      float ox = planes[node*6+0], oy = planes[node*6+1], oz = planes[node*6+2];
      float nx = planes[node*6+3], ny = planes[node*6+4], nz = planes[node*6+5];
      bool zero_n = (fabsf(nx) <= 1e-8f) && (fabsf(ny) <= 1e-8f) && (fabsf(nz) <= 1e-8f);
      if (zero_n) { nx = 1.0f; ny = 1.0f; nz = 1.0f; }
      float cos_nop = zero_n ? 0.0f : (nx*(ox-px) + ny*(oy-py) + nz*(oz-pz));

      bool zero_r = (fabsf(rx) <= 1e-8f) && (fabsf(ry) <= 1e-8f) && (fabsf(rz) <= 1e-8f);
      float sx = zero_r ? 1.0f : rx, sy = zero_r ? 1.0f : ry, sz = zero_r ? 1.0f : rz;
      float ss = sx*sx + sy*sy + sz*sz;
      float isn = rsqrtf((ss == 0.0f) ? 1.0f : ss);
      float denom = (nx*sx + ny*sy + nz*sz) * isn;
      float len_plane = zero_r ? 0.0f : (cos_nop / denom);
      float ls = lengths0[node];
      L = (ls != 0.0f) ? ls : len_plane;

      float rr = rx*rx + ry*ry + rz*rz;
      float ir = rsqrtf((rr == 0.0f) ? 1.0f : rr);
      px += L * rx * ir; py += L * ry * ir; pz += L * rz * ir;
    }

    if (i < N_STEPS - 1) {       // trail forces from steps 0..14
      float tf = sqrtf(rx*rx + ry*ry + rz*rz) * ((L < 0.0f) ? -1.0f : 1.0f);
      int e2 = seq_edges[i * N_TRAILS + s];
      out_forces[e2] = tf;
    } else if (node >= 0) {      // reactions = final residuals at last step
      out_react[node*3+0] = rx; out_react[node*3+1] = ry; out_react[node*3+2] = rz;
    }
  }
  __syncthreads();

  // ---- final xyz + edge lengths |xyz[v]-xyz[u]| straight from LDS ----
  for (int idx = s; idx < N_NODES * 3; idx += N_TRAILS) out_xyz[idx] = xyz_s[idx];
  for (int e = s; e < N_EDGES; e += N_TRAILS) {
    int ue = eu[e], ve = ev[e];
    float dx = xyz_s[ve*3+0] - xyz_s[ue*3+0];
    float dy = xyz_s[ve*3+1] - xyz_s[ue*3+1];
    float dz = xyz_s[ve*3+2] - xyz_s[ue*3+2];
    out_len[e] = sqrtf(dx*dx + dy*dy + dz*dz);
  }
}

// ---------------------------------------------------------------------------
extern "C" void kernel_launch(void* const* d_in, const int* in_sizes, int n_in,
                              void* d_out, int out_size, void* d_ws, size_t ws_size,
                              hipStream_t stream) {
  const float* xyz0      = (const float*)d_in[0];
  const float* loads     = (const float*)d_in[1];
  const float* lengths0  = (const float*)d_in[2];
  const float* planes    = (const float*)d_in[3];
  const float* forces0   = (const float*)d_in[4];
  const float* C         = (const float*)d_in[5];
  // d_in[6] (incidence, 130MB) is never read: inc == -C sign structure.
  const float* devmask   = (const float*)d_in[7];
  const int*   seqs      = (const int*)d_in[8];
  const int*   seq_edges = (const int*)d_in[9];

  int* eu = (int*)d_ws;
  int* ev = eu + N_EDGES;

  float* out        = (float*)d_out;
  float* out_xyz    = out;            // 12288
  float* out_react  = out + 12288;    // 12288
  float* out_len    = out + 24576;    //  7920
  float* out_loads  = out + 32496;    // 12288
  float* out_forces = out + 44784;    //  7920   (total 52704)

  cem_edges_wmma<<<dim3(N_EDGES / 16), dim3(32), 0, stream>>>(C, eu, ev);
  cem_scan<<<dim3(1), dim3(256), 0, stream>>>(xyz0, loads, lengths0, planes,
                                              forces0, devmask, seqs, seq_edges,
                                              eu, ev, out_xyz, out_react,
                                              out_len, out_loads, out_forces);
}